// Qwen3Attention_9603546874451
// MI455X (gfx1250) — compile-verified
//
#include <hip/hip_runtime.h>
#include <math.h>

#define B_   32
#define HID_ 1024
#define NH_  16
#define NKV_ 8
#define HD_  128
#define LC_  4096
#define R_   2   // NH/NKV

typedef float __attribute__((ext_vector_type(2))) v2f;
typedef float __attribute__((ext_vector_type(4))) v4f;
typedef float __attribute__((ext_vector_type(8))) v8f;

// ---------------------------------------------------------------------------
// C[M,N] = A[M,K] @ W[K,N], fp32 row-major, via V_WMMA_F32_16X16X4_F32.
// One wave computes a 32x64 block of C (MTILE=2 x NTILE=4 tiles of 16x16):
//  - A fragments are reused across the 4 N-tiles,
//  - B fragments are reused across the 2 M-tiles,
// so each k-step is 2 b64 + 8 b32 loads feeding 8 WMMAs (1.25 loads/wmma).
// All fragments of a k-step are loaded into distinct registers BEFORE the
// WMMA chain so the scheduler can stagger s_wait_loadcnt and overlap the
// next k-step's loads with the current WMMAs (avoids the v56/v57 register
// recycling that serialized round 2's inner loop).
// Fragment layouts per CDNA5 ISA 7.12.2:
//   A (16x4, 2 VGPR): lane l -> row = l%16; lanes 0-15 hold K={0,1},
//                     lanes 16-31 hold K={2,3}.
//   B (4x16, 2 VGPR): lane l -> col = l%16; same K split by lane half.
//   C/D (16x16, 8 VGPR): VGPR i, lanes 0-15 -> row i; lanes 16-31 -> row i+8.
// ---------------------------------------------------------------------------
#define MTILE 2
#define NTILE 4
__global__ __launch_bounds__(32)
void gemm_f32_wmma(const float* __restrict__ A, const float* __restrict__ W,
                   float* __restrict__ C, int M, int N, int K) {
  const int nt0  = blockIdx.x * NTILE;  // first N-tile of this wave's block
  const int mt0  = blockIdx.y * MTILE;  // first M-tile of this wave's block
  const int lane = threadIdx.x;
  const int half = lane >> 4;
  const int lm   = lane & 15;

  const int bcol0 = nt0 * 16 + lm;      // this lane's column in N-tile 0

  const float* Arow[MTILE];
  #pragma unroll
  for (int i = 0; i < MTILE; ++i)
    Arow[i] = A + (size_t)((mt0 + i) * 16 + lm) * K;

  v8f c[MTILE][NTILE] = {};

  #pragma unroll 2
  for (int k0 = 0; k0 < K; k0 += 4) {
    const int ka = k0 + 2 * half;       // lane's K-pair base

    // ---- load ALL fragments for this k-step first (distinct registers) ----
    v2f a[MTILE];
    #pragma unroll
    for (int i = 0; i < MTILE; ++i) {
      const float2 t = *(const float2*)(Arow[i] + ka);  // 8B-aligned
      a[i].x = t.x; a[i].y = t.y;
    }
    const float* W0 = W + (size_t)ka * N + bcol0;
    const float* W1 = W0 + N;
    v2f b[NTILE];
    #pragma unroll
    for (int j = 0; j < NTILE; ++j) {
      b[j].x = W0[j * 16];
      b[j].y = W1[j * 16];
    }

    // ---- then the 8-WMMA chain ----
    #pragma unroll
    for (int i = 0; i < MTILE; ++i)
      #pragma unroll
      for (int j = 0; j < NTILE; ++j)
        c[i][j] = __builtin_amdgcn_wmma_f32_16x16x4_f32(
            false, a[i], false, b[j], (short)0, c[i][j], false, false);
  }

  #pragma unroll
  for (int i = 0; i < MTILE; ++i) {
    #pragma unroll
    for (int j = 0; j < NTILE; ++j) {
      float* Cc = C + (size_t)((mt0 + i) * 16 + 8 * half) * N + bcol0 + j * 16;
      #pragma unroll
      for (int v = 0; v < 8; ++v)
        Cc[(size_t)v * N] = c[i][j][v];
    }
  }
}

// ---------------------------------------------------------------------------
// Per-(batch, head) RMSNorm + RoPE, in place. Heads [0,NH) = q rows in qbuf,
// heads [NH, NH+NKV) = new-k rows in kbuf. pos[b] gives the RoPE position.
// ---------------------------------------------------------------------------
__global__ __launch_bounds__(128)
void norm_rope(float* __restrict__ qbuf, float* __restrict__ kbuf,
               const float* __restrict__ qw, const float* __restrict__ kw,
               const int* __restrict__ pos) {
  const int bh = blockIdx.x;
  const int b  = bh / (NH_ + NKV_);
  const int h  = bh % (NH_ + NKV_);
  const int d  = threadIdx.x;        // 0..127

  float* vec;
  const float* w;
  if (h < NH_) { vec = qbuf + ((size_t)b * NH_  + h)          * HD_; w = qw; }
  else         { vec = kbuf + ((size_t)b * NKV_ + (h - NH_))  * HD_; w = kw; }

  __shared__ float sh[HD_];
  __shared__ float red[HD_];

  const float x = vec[d];
  red[d] = x * x;
  __syncthreads();
  for (int off = 64; off > 0; off >>= 1) {
    if (d < off) red[d] += red[d + off];
    __syncthreads();
  }
  const float rms = rsqrtf(red[0] * (1.0f / (float)HD_) + 1e-6f);
  sh[d] = w[d] * x * rms;
  __syncthreads();

  if (d < 64) {
    const float p  = (float)pos[b];
    // theta_d = THETA^(-2d/HD) = exp(-d * ln(1e6)/64)
    const float th = __expf(-(float)d * (13.815510557964274f / 64.0f));
    float sn, cs;
    __sincosf(p * th, &sn, &cs);
    const float x1 = sh[d], x2 = sh[d + 64];
    vec[d]      = x1 * cs - x2 * sn;
    vec[d + 64] = x2 * cs + x1 * sn;
  }
}

// ---------------------------------------------------------------------------
// Flash-style GQA decode. One block per (b, kv-head g); 8 waves stride over
// L = LC+1 rows (cache + new token). Each wave-lane holds 4 of the 128 dims
// (coalesced v4f / b128 row loads, non-temporal: 1 GB stream >> 192 MB L2).
// The wave's next row (+4 KB stride) is software-prefetched
// (global_prefetch_b8). Per-head online softmax; cross-wave LDS combine.
// ---------------------------------------------------------------------------
__global__ __launch_bounds__(256)
void attn_decode(const float* __restrict__ q,     // (B,NH,HD) normed+roped
                 const float* __restrict__ knew,  // (B,NKV,HD)
                 const float* __restrict__ vnew,  // (B,NKV,HD)
                 const float* __restrict__ kc,    // (B,NKV,LC,HD)
                 const float* __restrict__ vc,    // (B,NKV,LC,HD)
                 float* __restrict__ y) {         // (B,NH,HD)
  const int bg   = blockIdx.x;
  const int b    = bg / NKV_;
  const int g    = bg % NKV_;
  const int t    = threadIdx.x;
  const int w    = t >> 5;     // wave 0..7
  const int lane = t & 31;
  const int d0   = lane * 4;

  const float scale = 0.08838834764831845f;  // 1/sqrt(128)

  v4f qr[2];
  #pragma unroll
  for (int r = 0; r < R_; ++r)
    qr[r] = *(const v4f*)(q + ((size_t)b * NH_ + g * R_ + r) * HD_ + d0);

  const float* Kb = kc   + ((size_t)b * NKV_ + g) * (size_t)LC_ * HD_;
  const float* Vb = vc   + ((size_t)b * NKV_ + g) * (size_t)LC_ * HD_;
  const float* kn = knew + ((size_t)b * NKV_ + g) * HD_;
  const float* vn = vnew + ((size_t)b * NKV_ + g) * HD_;

  float m[R_] = {-INFINITY, -INFINITY};
  float s[R_] = {0.f, 0.f};
  v4f acc[R_] = {{0.f, 0.f, 0.f, 0.f}, {0.f, 0.f, 0.f, 0.f}};

  const int L = LC_ + 1;
  for (int l = w; l < L; l += 8) {
    v4f kv, vv;
    if (l < LC_) {
      kv = __builtin_nontemporal_load((const v4f*)(Kb + (size_t)l * HD_ + d0));
      vv = __builtin_nontemporal_load((const v4f*)(Vb + (size_t)l * HD_ + d0));
      if (l + 8 < LC_) {   // prefetch this wave's next rows (stride 4 KB)
        __builtin_prefetch(Kb + (size_t)(l + 8) * HD_ + d0, 0, 0);
        __builtin_prefetch(Vb + (size_t)(l + 8) * HD_ + d0, 0, 0);
      }
    } else {
      kv = *(const v4f*)(kn + d0);
      vv = *(const v4f*)(vn + d0);
    }
    #pragma unroll
    for (int r = 0; r < R_; ++r) {
      float dp = kv.x * qr[r].x + kv.y * qr[r].y +
                 kv.z * qr[r].z + kv.w * qr[r].w;
      for (int off = 16; off > 0; off >>= 1)      // wave32 all-reduce
        dp += __shfl_xor(dp, off, 32);
      const float sc = dp * scale;
      const float mn = fmaxf(m[r], sc);
      const float f  = __expf(m[r] - mn);         // exp(-inf)=0 on first hit
      const float p  = __expf(sc   - mn);
      s[r]   = s[r] * f + p;
      acc[r] = acc[r] * f + p * vv;
      m[r]   = mn;
    }
  }

  __shared__ float sm[8][R_];
  __shared__ float ss[8][R_];
  __shared__ float sacc[8][R_][HD_];
  if (lane == 0) {
    #pragma unroll
    for (int r = 0; r < R_; ++r) { sm[w][r] = m[r]; ss[w][r] = s[r]; }
  }
  #pragma unroll
  for (int r = 0; r < R_; ++r)
    *(v4f*)(&sacc[w][r][d0]) = acc[r];
  __syncthreads();

  // 256 threads -> R_ heads x 128 dims
  const int r = t >> 7;
  const int d = t & (HD_ - 1);
  float M = -INFINITY;
  #pragma unroll
  for (int i = 0; i < 8; ++i) M = fmaxf(M, sm[i][r]);
  float S = 0.f, o = 0.f;
  #pragma unroll
  for (int i = 0; i < 8; ++i) {
    const float f = __expf(sm[i][r] - M);
    S += ss[i][r] * f;
    o += sacc[i][r][d] * f;
  }
  y[((size_t)b * NH_ + g * R_ + r) * HD_ + d] = o / S;
}

// ---------------------------------------------------------------------------
extern "C" void kernel_launch(void* const* d_in, const int* in_sizes, int n_in,
                              void* d_out, int out_size, void* d_ws, size_t ws_size,
                              hipStream_t stream) {
  (void)in_sizes; (void)n_in; (void)out_size; (void)ws_size;

  const float* x   = (const float*)d_in[0];  // (B,1,HID)
  const float* Wq  = (const float*)d_in[1];  // (HID, NH*HD)
  const float* Wk  = (const float*)d_in[2];  // (HID, NKV*HD)
  const float* Wv  = (const float*)d_in[3];  // (HID, NKV*HD)
  const float* Wo  = (const float*)d_in[4];  // (NH*HD, HID)
  const float* qw  = (const float*)d_in[5];  // (HD,)
  const float* kw  = (const float*)d_in[6];  // (HD,)
  const float* kc  = (const float*)d_in[7];  // (B,NKV,LC,HD)
  const float* vc  = (const float*)d_in[8];  // (B,NKV,LC,HD)
  const int*   pos = (const int*)d_in[9];    // (B,)
  float* out = (float*)d_out;                // (B,1,HID)

  // Workspace layout (fp32): q | k_new | v_new | y  => 768 KB total
  float* ws   = (float*)d_ws;
  float* qraw = ws;                                 // B*NH*HD
  float* knew = qraw + (size_t)B_ * NH_  * HD_;     // B*NKV*HD
  float* vnew = knew + (size_t)B_ * NKV_ * HD_;     // B*NKV*HD
  float* ybuf = vnew + (size_t)B_ * NKV_ * HD_;     // B*NH*HD

  // 1) QKV projections (fp32 WMMA GEMM, 32x64 block per wave)
  gemm_f32_wmma<<<dim3((NH_  * HD_) / (16 * NTILE), B_ / (16 * MTILE)),
                  32, 0, stream>>>(x, Wq, qraw, B_, NH_ * HD_, HID_);
  gemm_f32_wmma<<<dim3((NKV_ * HD_) / (16 * NTILE), B_ / (16 * MTILE)),
                  32, 0, stream>>>(x, Wk, knew, B_, NKV_ * HD_, HID_);
  gemm_f32_wmma<<<dim3((NKV_ * HD_) / (16 * NTILE), B_ / (16 * MTILE)),
                  32, 0, stream>>>(x, Wv, vnew, B_, NKV_ * HD_, HID_);

  // 2) RMSNorm + RoPE on q and new k (in place)
  norm_rope<<<B_ * (NH_ + NKV_), 128, 0, stream>>>(qraw, knew, qw, kw, pos);

  // 3) Flash decode over LC+1 positions (HBM-bound: ~1.07 GB of K/V)
  attn_decode<<<B_ * NKV_, 256, 0, stream>>>(qraw, knew, vnew, kc, vc, ybuf);

  // 4) Output projection: 32 x 1024 x 2048 -> d_out
  gemm_f32_wmma<<<dim3(HID_ / (16 * NTILE), B_ / (16 * MTILE)),
                  32, 0, stream>>>(ybuf, Wo, out, B_, HID_, NH_ * HD_);
}